// ConnectLoss_40003325395659
// MI455X (gfx1250) — compile-verified
//
#include <hip/hip_runtime.h>

#define KCH   17
#define PER_B 589824      // 768*768 elements per batch image
#define M_TOT 2359296     // 4*768*768
#define NG8   (M_TOT / 8) // 8-element groups for the main kernel
#define EPSF  1e-7f
#define LN2F  0.6931471805599453f

typedef __attribute__((ext_vector_type(2))) float v2f;
typedef __attribute__((ext_vector_type(4))) float v4f;
typedef __attribute__((ext_vector_type(8))) float v8f;

// ---- workspace layout (doubles). log-quantities stored in log2 units. ----
#define WS_A    0     // [256]  A[n][k]  = sum_{t==n+1} log2(p)-log2(1-p) for ch k+1
#define WS_TP   256   // [256]  TP[n][k] = sum_{t==n+1} P[k+1]
#define WS_LOGS 512   // [17]   sum_m log2(1-p[k])
#define WS_PSUM 529   // [17]   sum_m P[k]
#define WS_CNT  546   // [17]   count(t==n)
#define WS_D00  563   // background sum_{t==0} log2(p0)-log2(1-p0)
#define WS_TP00 564   // background sum_{t==0} P[0]
#define WS_TOTAL 565

__global__ void zero_ws(double* ws) {
    int i = threadIdx.x;
    if (i < WS_TOTAL) ws[i] = 0.0;
}

__device__ __forceinline__ float clipf(float x) {
    return fminf(fmaxf(x, EPSF), 1.0f - EPSF);
}

// ---------------- main: instance channels 1..16 vs targets 1..16 ----------------
__global__ __launch_bounds__(256)
void pair_reduce(const float* __restrict__ pred,   // [4,17,768,768]
                 const int* __restrict__ tgt,      // [4,1,768,768]
                 double* __restrict__ ws) {
    const int tid  = threadIdx.x;
    const int lane = tid & 31;
    const int h    = lane >> 4;      // lane-half: which K-slot pair this lane feeds
    const int col  = lane & 15;      // B column / A row
    const int ch   = col + 1;        // prediction channel streamed by this lane
    const int nid  = col + 1;        // target id matched by this lane's A row

    const long wave   = (long)blockIdx.x * (blockDim.x >> 5) + (tid >> 5);
    const long nwaves = (long)gridDim.x * (blockDim.x >> 5);

    v8f acc_d  = {};   // one-hot(T) x (log2 p - log2(1-p))
    v8f acc_tp = {};   // one-hot(T) x P
    float s_l2m = 0.f, s_p = 0.f, s_cnt = 0.f;

    for (long g = wave; g < NG8; g += nwaves) {      // wave-uniform trip count
        const long m_base = g * 8;                   // 8-aligned: never straddles batch b
        const int  b = (int)(m_base / PER_B);
        const long r = m_base - (long)b * PER_B + (h << 2);
        const long m = m_base + (h << 2);

        int4 tt = *(const int4*)(tgt + m);
        v4f  pv = *(const v4f*)(pred + ((long)b * KCH + ch) * PER_B + r);

        float pc0 = clipf(pv.x), pc1 = clipf(pv.y);
        float pc2 = clipf(pv.z), pc3 = clipf(pv.w);
        // raw v_log_f32 (base-2); ln2 scaling deferred to finalize
        float l0 = __builtin_amdgcn_logf(1.0f - pc0);
        float l1 = __builtin_amdgcn_logf(1.0f - pc1);
        float l2 = __builtin_amdgcn_logf(1.0f - pc2);
        float l3 = __builtin_amdgcn_logf(1.0f - pc3);
        float d0 = __builtin_amdgcn_logf(pc0) - l0;
        float d1 = __builtin_amdgcn_logf(pc1) - l1;
        float d2 = __builtin_amdgcn_logf(pc2) - l2;
        float d3 = __builtin_amdgcn_logf(pc3) - l3;

        float a0 = (tt.x == nid) ? 1.0f : 0.0f;
        float a1 = (tt.y == nid) ? 1.0f : 0.0f;
        float a2 = (tt.z == nid) ? 1.0f : 0.0f;
        float a3 = (tt.w == nid) ? 1.0f : 0.0f;

        v2f A01 = {a0, a1}, A23 = {a2, a3};
        v2f D01 = {d0, d1}, D23 = {d2, d3};
        v2f P01 = {pv.x, pv.y}, P23 = {pv.z, pv.w};

        // K-slot assignment is arbitrary for a pure sum: two WMMAs cover all 8 elements
        acc_d  = __builtin_amdgcn_wmma_f32_16x16x4_f32(false, A01, false, D01, (short)0, acc_d,  false, false);
        acc_d  = __builtin_amdgcn_wmma_f32_16x16x4_f32(false, A23, false, D23, (short)0, acc_d,  false, false);
        acc_tp = __builtin_amdgcn_wmma_f32_16x16x4_f32(false, A01, false, P01, (short)0, acc_tp, false, false);
        acc_tp = __builtin_amdgcn_wmma_f32_16x16x4_f32(false, A23, false, P23, (short)0, acc_tp, false, false);

        s_l2m += (l0 + l1) + (l2 + l3);
        s_p   += (pv.x + pv.y) + (pv.z + pv.w);
        s_cnt += (a0 + a1) + (a2 + a3);
    }

    // ---- block-level LDS reduction (ds_add_f64), then one global atomic per slot ----
    __shared__ double sh[560];                 // [0,512): A|TP tiles, [512,560): logs|psum|cnt
    for (int i = tid; i < 560; i += 256) sh[i] = 0.0;
    __syncthreads();

#pragma unroll
    for (int j = 0; j < 8; ++j) {              // C/D layout: VGPR j, lane -> row j+8h, col
        int row = j + 8 * h;
        atomicAdd(&sh[row * 16 + col],       (double)acc_d[j]);
        atomicAdd(&sh[256 + row * 16 + col], (double)acc_tp[j]);
    }
    atomicAdd(&sh[512 + col], (double)s_l2m);
    atomicAdd(&sh[528 + col], (double)s_p);
    atomicAdd(&sh[544 + col], (double)s_cnt);
    __syncthreads();

    for (int i = tid; i < 560; i += 256) {
        int gi;
        if      (i < 512) gi = i;                          // WS_A | WS_TP contiguous
        else if (i < 528) gi = WS_LOGS + 1 + (i - 512);
        else if (i < 544) gi = WS_PSUM + 1 + (i - 528);
        else              gi = WS_CNT  + 1 + (i - 544);
        atomicAdd(&ws[gi], sh[i]);
    }
}

// ---------------- background: channel 0 vs (t == 0), full-wave parallel ----------------
__global__ __launch_bounds__(256)
void bg_reduce(const float* __restrict__ pred,
               const int* __restrict__ tgt,
               double* __restrict__ ws) {
    const long idx = (long)blockIdx.x * blockDim.x + threadIdx.x;
    const long nth = (long)gridDim.x * blockDim.x;
    float s_l2m = 0.f, s_p = 0.f, s_cnt = 0.f, s_d = 0.f, s_tp = 0.f;

    for (long i = idx; i < M_TOT / 4; i += nth) {
        const long m = i * 4;
        const int  b = (int)(m / PER_B);
        const long r = m - (long)b * PER_B;
        v4f  q  = *(const v4f*)(pred + ((long)b * KCH) * PER_B + r);
        int4 tt = *(const int4*)(tgt + m);

        float qc0 = clipf(q.x), qc1 = clipf(q.y), qc2 = clipf(q.z), qc3 = clipf(q.w);
        float l0 = __builtin_amdgcn_logf(1.0f - qc0);
        float l1 = __builtin_amdgcn_logf(1.0f - qc1);
        float l2 = __builtin_amdgcn_logf(1.0f - qc2);
        float l3 = __builtin_amdgcn_logf(1.0f - qc3);
        float d0 = __builtin_amdgcn_logf(qc0) - l0;
        float d1 = __builtin_amdgcn_logf(qc1) - l1;
        float d2 = __builtin_amdgcn_logf(qc2) - l2;
        float d3 = __builtin_amdgcn_logf(qc3) - l3;
        float z0 = (tt.x == 0) ? 1.0f : 0.0f;
        float z1 = (tt.y == 0) ? 1.0f : 0.0f;
        float z2 = (tt.z == 0) ? 1.0f : 0.0f;
        float z3 = (tt.w == 0) ? 1.0f : 0.0f;

        s_l2m += (l0 + l1) + (l2 + l3);
        s_p   += (q.x + q.y) + (q.z + q.w);
        s_cnt += (z0 + z1) + (z2 + z3);
        s_d   += (z0 * d0 + z1 * d1) + (z2 * d2 + z3 * d3);
        s_tp  += (z0 * q.x + z1 * q.y) + (z2 * q.z + z3 * q.w);
    }

    // wave32 shuffle reduction, lane 0 commits
#pragma unroll
    for (int off = 16; off > 0; off >>= 1) {
        s_l2m += __shfl_down(s_l2m, off, 32);
        s_p   += __shfl_down(s_p,   off, 32);
        s_cnt += __shfl_down(s_cnt, off, 32);
        s_d   += __shfl_down(s_d,   off, 32);
        s_tp  += __shfl_down(s_tp,  off, 32);
    }
    if ((threadIdx.x & 31) == 0) {
        atomicAdd(&ws[WS_LOGS], (double)s_l2m);
        atomicAdd(&ws[WS_PSUM], (double)s_p);
        atomicAdd(&ws[WS_CNT],  (double)s_cnt);
        atomicAdd(&ws[WS_D00],  (double)s_d);
        atomicAdd(&ws[WS_TP00], (double)s_tp);
    }
}

// ---------------- tiny epilogue: L matrix + sequential greedy assignment ----------------
__global__ void finalize(const double* __restrict__ ws, float* __restrict__ out) {
    __shared__ float Ls[256];
    const double Mf = (double)M_TOT;
    int tid = threadIdx.x;
    int n = tid >> 4, k = tid & 15;
    // log sums are in log2 units -> scale by ln2 here
    float mean_l = (float)(ws[WS_LOGS + k + 1] / Mf) * LN2F;
    float bce  = -(float)(ws[WS_A + n * 16 + k] / Mf) * LN2F - mean_l;
    float tp   = (float)ws[WS_TP + n * 16 + k];
    float dice = 1.0f - (2.0f * tp + EPSF) /
                 ((float)ws[WS_CNT + n + 1] + (float)ws[WS_PSUM + k + 1] + EPSF);
    Ls[tid] = bce + dice;
    __syncthreads();
    if (tid == 0) {
        float bce0  = -((float)(ws[WS_D00] / Mf) + (float)(ws[WS_LOGS] / Mf)) * LN2F;
        float dice0 = 1.0f - (2.0f * (float)ws[WS_TP00] + EPSF) /
                      ((float)ws[WS_CNT] + (float)ws[WS_PSUM] + EPSF);
        float tot = bce0 + dice0;
        bool avail[16];
        for (int i = 0; i < 16; ++i) avail[i] = true;
        for (int nn = 0; nn < 16; ++nn) {            // sequential greedy, first-index ties
            float best = __builtin_inff(); int kb = 0;
            for (int kk = 0; kk < 16; ++kk) {
                float v = Ls[nn * 16 + kk];
                if (avail[kk] && v < best) { best = v; kb = kk; }
            }
            avail[kb] = false;
            tot += best;
        }
        out[0] = tot / 16.0f;
    }
}

extern "C" void kernel_launch(void* const* d_in, const int* in_sizes, int n_in,
                              void* d_out, int out_size, void* d_ws, size_t ws_size,
                              hipStream_t stream) {
    const float* pred = (const float*)d_in[0];
    const int*   tgt  = (const int*)d_in[1];
    double* ws = (double*)d_ws;
    float* outf = (float*)d_out;

    hipLaunchKernelGGL(zero_ws, dim3(1), dim3(1024), 0, stream, ws);
    // 256 blocks x 8 waves = 2048 waves; 144 eight-element groups per wave
    hipLaunchKernelGGL(pair_reduce, dim3(256), dim3(256), 0, stream, pred, tgt, ws);
    hipLaunchKernelGGL(bg_reduce, dim3(96), dim3(256), 0, stream, pred, tgt, ws);
    hipLaunchKernelGGL(finalize, dim3(1), dim3(256), 0, stream, ws, outf);
}